// EnhancedLinkPredictor_33681133535939
// MI455X (gfx1250) — compile-verified
//
#include <hip/hip_runtime.h>
#include <hip/hip_bf16.h>

typedef __attribute__((ext_vector_type(2))) float v2f;
typedef __attribute__((ext_vector_type(8))) float v8f;

#define N_D    8192
#define N_P    4096
#define E_DDI  131072
#define E_DP   131072
#define E_PRED 16384
#define HID    128
#define IN_DIM 256

#define DDI_WORDS 256   // 8192 bits / 32
#define DP_WORDS  128   // 4096 bits / 32

// ---------------------------------------------------------------------------
// 1) Build symmetric drug-drug bitset adjacency: one thread per directed edge.
// ---------------------------------------------------------------------------
__global__ void build_ddi_kernel(const int* __restrict__ ei, unsigned* __restrict__ adj) {
    int t = blockIdx.x * blockDim.x + threadIdx.x;
    if (t >= 2 * E_DDI) return;
    int a, b;
    if (t < E_DDI) { a = ei[t];          b = ei[E_DDI + t]; }
    else           { int u = t - E_DDI; a = ei[E_DDI + u]; b = ei[u]; }
    atomicOr(&adj[(size_t)a * DDI_WORDS + (b >> 5)], 1u << (b & 31));
}

__global__ void build_dp_kernel(const int* __restrict__ ei, unsigned* __restrict__ adj) {
    int t = blockIdx.x * blockDim.x + threadIdx.x;
    if (t >= E_DP) return;
    int dr = ei[t];
    int pr = ei[E_DP + t];
    atomicOr(&adj[(size_t)dr * DP_WORDS + (pr >> 5)], 1u << (pr & 31));
}

// ---------------------------------------------------------------------------
// 2) Gather z_drug[src], z_drug[dst] into pair[:, 0:64] and pair[:, 64:128].
//    Coalesced: 64 consecutive threads handle one edge's 64 features.
// ---------------------------------------------------------------------------
__global__ void gather_pair_kernel(const float* __restrict__ z_drug,
                                   const int* __restrict__ pred,
                                   float* __restrict__ pair) {
    int t = blockIdx.x * blockDim.x + threadIdx.x;   // E_PRED * 64 threads
    int e = t >> 6;
    int f = t & 63;
    int s = pred[e];
    int d = pred[E_PRED + e];
    pair[(size_t)e * IN_DIM + f]      = z_drug[(size_t)s * 64 + f];
    pair[(size_t)e * IN_DIM + 64 + f] = z_drug[(size_t)d * 64 + f];
}

// ---------------------------------------------------------------------------
// 3) Common-neighbor mean pooling over a bitset adjacency.
//    One thread per prediction edge; 64 register accumulators.
//    Bitsets are L2-resident (8MB/4MB << 192MB L2).
// ---------------------------------------------------------------------------
__global__ void cn_mean_kernel(const unsigned* __restrict__ adj, int words,
                               const float* __restrict__ feat,
                               const int* __restrict__ pred,
                               float* __restrict__ pair, int out_off) {
    int e = blockIdx.x * blockDim.x + threadIdx.x;
    if (e >= E_PRED) return;
    int s = pred[e];
    int d = pred[E_PRED + e];
    const unsigned* rs = adj + (size_t)s * words;
    const unsigned* rd = adj + (size_t)d * words;

    float acc[64];
#pragma unroll
    for (int f = 0; f < 64; ++f) acc[f] = 0.0f;

    int cnt = 0;
    for (int w = 0; w < words; ++w) {
        unsigned u = rs[w] & rd[w];
        cnt += __popc(u);
        while (u) {
            int j = __builtin_ctz(u);
            const float* zr = feat + (size_t)(w * 32 + j) * 64;
#pragma unroll
            for (int f = 0; f < 64; ++f) acc[f] += zr[f];
            u &= u - 1;
        }
    }
    float inv = 1.0f / (float)(cnt > 1 ? cnt : 1);
    float* o = pair + (size_t)e * IN_DIM + out_off;
#pragma unroll
    for (int f = 0; f < 64; ++f) o[f] = acc[f] * inv;
}

// ---------------------------------------------------------------------------
// 4) Fused MLP:  logit = relu(pair @ W1 + b1) @ W2 + b2 ; out = sigmoid(logit)
//    Block = 256 threads = 8 wave32 waves; block owns 16 rows (edges).
//    Wave w computes hidden columns [16w, 16w+16) with V_WMMA_F32_16X16X4_F32
//    over K=256 (64 WMMA issues), then fuses bias/ReLU/W2-dot into an LDS
//    logit accumulator. Lanes 0..15 of the block emit the sigmoid.
//
//    F32 WMMA operand layout (ISA 7.12.2, 32-bit 16x4 A / 16x16 C):
//      A  : lane m=lane&15, VGPR j holds K = 2*(lane>>4) + j
//      B  : lane n=lane&15, VGPR j holds K = 2*(lane>>4) + j
//      C/D: VGPR r holds (M = r + 8*(lane>>4), N = lane&15)
// ---------------------------------------------------------------------------
__global__ void mlp_wmma_kernel(const float* __restrict__ pair,
                                const float* __restrict__ W1,
                                const float* __restrict__ b1,
                                const float* __restrict__ W2,
                                const float* __restrict__ b2,
                                float* __restrict__ out) {
    const int block_m = blockIdx.x * 16;          // 16 edges per block
    const int lane    = threadIdx.x & 31;
    const int wave    = threadIdx.x >> 5;         // 0..7
    const int nb      = wave * 16;                // hidden-column tile base
    const int lm      = lane & 15;                // row within tile (A) or col (B/C)
    const int kh      = (lane >> 4) * 2;          // K sub-offset for this lane half
    const int mh      = (lane >> 4) * 8;          // C/D row offset for this lane half

    __shared__ float logit_acc[16];
    if (threadIdx.x < 16) logit_acc[threadIdx.x] = 0.0f;
    __syncthreads();

    const float* arow = pair + (size_t)(block_m + lm) * IN_DIM;

    v8f c = {};
    // K = 256, step 4 -> 64 wmma issues per wave. EXEC is all-1s (no divergence).
#pragma unroll 4
    for (int k = 0; k < IN_DIM; k += 4) {
        v2f a, b;
        a.x = arow[k + kh + 0];
        a.y = arow[k + kh + 1];
        b.x = W1[(size_t)(k + kh + 0) * HID + nb + lm];
        b.y = W1[(size_t)(k + kh + 1) * HID + nb + lm];
        c = __builtin_amdgcn_wmma_f32_16x16x4_f32(
                /*neg_a=*/false, a, /*neg_b=*/false, b,
                /*c_mod=*/(short)0, c, /*reuse_a=*/false, /*reuse_b=*/false);
    }

    // Fused epilogue: h = relu(c + b1[n]);  partial logit += h * W2[n]
    const float bias = b1[nb + lm];
    const float w2v  = W2[nb + lm];
#pragma unroll
    for (int r = 0; r < 8; ++r) {
        float h = c[r] + bias;
        h = h > 0.0f ? h : 0.0f;
        atomicAdd(&logit_acc[r + mh], h * w2v);
    }
    __syncthreads();

    if (threadIdx.x < 16) {
        float l = logit_acc[threadIdx.x] + b2[0];
        out[block_m + threadIdx.x] = 1.0f / (1.0f + __expf(-l));
    }
}

// ---------------------------------------------------------------------------
// Launch
// ---------------------------------------------------------------------------
extern "C" void kernel_launch(void* const* d_in, const int* in_sizes, int n_in,
                              void* d_out, int out_size, void* d_ws, size_t ws_size,
                              hipStream_t stream) {
    const float* z_drug = (const float*)d_in[0];
    const float* z_prot = (const float*)d_in[1];
    const int*   ddi_ei = (const int*)d_in[2];
    const int*   dp_ei  = (const int*)d_in[3];
    const int*   pred   = (const int*)d_in[4];
    const float* W1     = (const float*)d_in[5];
    const float* b1     = (const float*)d_in[6];
    const float* W2     = (const float*)d_in[7];
    const float* b2     = (const float*)d_in[8];
    float* out = (float*)d_out;

    // Workspace layout:
    //   [0, 8MB)        adj_ddi bitset  (8192 rows x 256 u32)
    //   [8MB, 12MB)     adj_dp bitset   (8192 rows x 128 u32)
    //   [12MB, 28MB)    pair features   (16384 x 256 f32)
    char* ws = (char*)d_ws;
    unsigned* adj_ddi = (unsigned*)ws;
    unsigned* adj_dp  = (unsigned*)(ws + (8ull << 20));
    float*    pair    = (float*)(ws + (12ull << 20));

    hipMemsetAsync(adj_ddi, 0, (12ull << 20), stream);  // both bitsets

    build_ddi_kernel<<<(2 * E_DDI + 255) / 256, 256, 0, stream>>>(ddi_ei, adj_ddi);
    build_dp_kernel<<<(E_DP + 255) / 256, 256, 0, stream>>>(dp_ei, adj_dp);

    gather_pair_kernel<<<(E_PRED * 64) / 256, 256, 0, stream>>>(z_drug, pred, pair);

    cn_mean_kernel<<<(E_PRED + 255) / 256, 256, 0, stream>>>(
        adj_ddi, DDI_WORDS, z_drug, pred, pair, 128);
    cn_mean_kernel<<<(E_PRED + 255) / 256, 256, 0, stream>>>(
        adj_dp, DP_WORDS, z_prot, pred, pair, 192);

    mlp_wmma_kernel<<<E_PRED / 16, 256, 0, stream>>>(pair, W1, b1, W2, b2, out);
}